// Alignment_35424890257770
// MI455X (gfx1250) — compile-verified
//
#include <hip/hip_runtime.h>
#include <hip/hip_bf16.h>

// ---------------------------------------------------------------------------
// Shapes (fixed by reference): B=64, hat_L=8, N=16, d=32
// Z_s/Z_t: [B, L, N, d]  (flat stride: b*4096 + t*512 + n*32 + k)
// E_s/E_t: [B, L, N, N]  (flat stride: b*2048 + t*256 + m*16 + n)
//
// CORAL trick: ||cov_s - cov_t||_F^2 = (||Ks||^2 - 2||Kst||^2 + ||Kt||^2)/63^2
// with K* the 64x64 Gram matrices of the *centered* feature matrices, and the
// full-feature Gram = sum of per-layer Grams. All GEMM-shaped work runs on
// V_WMMA_F32_16X16X4_F32 (native fp32 matrix pipe; keeps reference precision).
// Gram operands are staged into LDS with CDNA5 async-to-LDS copies when the
// toolchain exposes them (ASYNCcnt + s_wait_asynccnt), else a sync LDS copy.
// ---------------------------------------------------------------------------

typedef __attribute__((ext_vector_type(2))) float v2f;
typedef __attribute__((ext_vector_type(4))) float v4f;
typedef __attribute__((ext_vector_type(8))) float v8f;

#define NB   64    // batch
#define NL   8     // hat_L
#define NN   16    // N
#define ND   32    // d
#define NF   512   // N*d  (per-layer feature count)
#define KS   128   // K-slice width for LDS staging
#define LROW 132   // padded LDS row stride (132 % 64(banks)==4 -> conflict-free)

#if defined(__has_builtin)
#  if __has_builtin(__builtin_amdgcn_global_load_async_to_lds_b128)
#    define USE_ASYNC_LDS 1
#  endif
#endif
#ifndef USE_ASYNC_LDS
#  define USE_ASYNC_LDS 0
#endif

#if USE_ASYNC_LDS
// Exact parameter types per hipcc diagnostic: pointer to 16B int vector,
// global (addrspace 1) source first, LDS (addrspace 3) destination second.
typedef int b128_t __attribute__((vector_size(16)));
typedef __attribute__((address_space(1))) b128_t gb128_t;
typedef __attribute__((address_space(3))) b128_t lb128_t;
#endif

static __device__ __forceinline__ v8f wmma_f32(v2f a, v2f b, v8f c) {
  // (neg_a, A, neg_b, B, c_mod, C, reuse_a, reuse_b)
  return __builtin_amdgcn_wmma_f32_16x16x4_f32(false, a, false, b, (short)0, c,
                                               false, false);
}

// ---------------------------------------------------------------------------
// 1) Center Z over the batch dim, write layer-major copies [L][B][512].
//    grid = 16 blocks ((tensor<<3)|layer), 256 threads.
// ---------------------------------------------------------------------------
__global__ void center_kernel(const float* __restrict__ Zs,
                              const float* __restrict__ Zt,
                              float* __restrict__ ZCs,
                              float* __restrict__ ZCt) {
  const int tensor = blockIdx.x >> 3;
  const int t = blockIdx.x & 7;
  const float* Z = tensor ? Zt : Zs;
  float* ZC = tensor ? ZCt : ZCs;
  for (int f = threadIdx.x; f < NF; f += blockDim.x) {
    const float* src = Z + t * NF + f;  // + b*4096
    float sum = 0.f;
#pragma unroll 8
    for (int b = 0; b < NB; ++b) sum += src[b * (NL * NF)];
    const float mean = sum * (1.0f / NB);
    float* dst = ZC + (t * NB) * NF + f;  // + b*512
#pragma unroll 8
    for (int b = 0; b < NB; ++b) dst[b * NF] = src[b * (NL * NF)] - mean;
  }
}

// ---------------------------------------------------------------------------
// 2) Per-layer 64x64 Grams G = X * Y^T, K=512, via WMMA f32 16x16x4.
//    grid = 24 blocks (layer*3 + {ss,tt,st}), 512 threads = 16 waves,
//    wave w computes output tile (mt = w>>2, nt = w&3).
//    K is processed in 4 slices of 128; each slice of X and Y is staged into
//    LDS (async-to-LDS when available) and WMMA fragments are fed from
//    ds_load_b64 with a bank-conflict-free padded row stride.
//    A fragment layout (16x4 fp32): lane&15 -> M row, lane>>4 -> K pair.
// ---------------------------------------------------------------------------
__global__ void gram_kernel(const float* __restrict__ ZCs,
                            const float* __restrict__ ZCt,
                            float* __restrict__ grams) {
  extern __shared__ float smem[];        // [2][64][LROW] = 67584 bytes
  float* Xl = smem;
  float* Yl = smem + NB * LROW;

  const int t = blockIdx.x / 3;
  const int g = blockIdx.x % 3;  // 0: s.s, 1: t.t, 2: s.t
  const float* X = ((g == 1) ? ZCt : ZCs) + t * NB * NF;
  const float* Y = ((g == 0) ? ZCs : ZCt) + t * NB * NF;

  const int wave = threadIdx.x >> 5;
  const int lane = threadIdx.x & 31;
  const int mt = wave >> 2, nt = wave & 3;
  const int l16 = lane & 15, lh = lane >> 4;

  v8f c = {};
  for (int ks = 0; ks < NF / KS; ++ks) {
    // ---- stage the 64 x 128 slices of X and Y into LDS ----
    for (int i = threadIdx.x; i < NB * (KS / 4); i += blockDim.x) {
      const int row = i >> 5;       // 32 float4 per row
      const int c4 = (i & 31) << 2; // float offset within row
      const float* gx = X + row * NF + ks * KS + c4;
      const float* gy = Y + row * NF + ks * KS + c4;
      float* lx = Xl + row * LROW + c4;  // row*528B -> 16B aligned
      float* ly = Yl + row * LROW + c4;
#if USE_ASYNC_LDS
      __builtin_amdgcn_global_load_async_to_lds_b128((gb128_t*)gx, (lb128_t*)lx,
                                                     0, 0);
      __builtin_amdgcn_global_load_async_to_lds_b128((gb128_t*)gy, (lb128_t*)ly,
                                                     0, 0);
#else
      *(v4f*)lx = *(const v4f*)gx;
      *(v4f*)ly = *(const v4f*)gy;
#endif
    }
#if USE_ASYNC_LDS
#  if __has_builtin(__builtin_amdgcn_s_wait_asynccnt)
    __builtin_amdgcn_s_wait_asynccnt(0);
#  else
    asm volatile("s_wait_asynccnt 0x0" ::: "memory");
#  endif
#endif
    __syncthreads();

    // ---- 32 WMMA K-chunks out of LDS ----
    const float* arow = Xl + (mt * 16 + l16) * LROW + lh * 2;
    const float* brow = Yl + (nt * 16 + l16) * LROW + lh * 2;
#pragma unroll 4
    for (int kc = 0; kc < KS / 4; ++kc) {
      v2f a = *(const v2f*)(arow + kc * 4);
      v2f b = *(const v2f*)(brow + kc * 4);
      c = wmma_f32(a, b, c);
    }
    __syncthreads();  // before next slice overwrites LDS
  }

  float* G = grams + (t * 3 + g) * (64 * 64);
  const int row0 = mt * 16 + lh * 8;
  const int col = nt * 16 + l16;
#pragma unroll
  for (int r = 0; r < 8; ++r) G[(row0 + r) * 64 + col] = c[r];
}

// ---------------------------------------------------------------------------
// 3) Node covariances C = Y^T Y / 63, Y = centered [64, 32] slice.
//    One wave per (tensor, t, n): 2x2 tiles of 32x32, K=64 -> 64 WMMAs.
//    A(frag for tile mt) == B(frag for tile mt) per lane, so two fragment
//    loads feed 4 WMMAs. Also reduces ||C||_F^2 -> ss/tt via lane shuffles.
//    grid = 256 blocks, 32 threads.
// ---------------------------------------------------------------------------
__global__ void cov_kernel(const float* __restrict__ ZCs,
                           const float* __restrict__ ZCt,
                           float* __restrict__ Cs, float* __restrict__ Ct,
                           float* __restrict__ ss, float* __restrict__ tt) {
  const int n = blockIdx.x & 15;
  const int t = (blockIdx.x >> 4) & 7;
  const int tensor = blockIdx.x >> 7;
  const float* Zc = (tensor ? ZCt : ZCs) + t * NB * NF + n * ND;
  float* C = (tensor ? Ct : Cs) + (t * NN + n) * (ND * ND);

  const int lane = threadIdx.x;
  const int l16 = lane & 15, lh = lane >> 4;

  v8f a00 = {}, a01 = {}, a10 = {}, a11 = {};
#pragma unroll 4
  for (int kc = 0; kc < NB / 4; ++kc) {
    const int k = kc * 4 + lh * 2;  // batch-row pair for this lane half
    const float* p = Zc + k * NF + l16;
    v2f f0, f1;
    f0[0] = p[0];   f0[1] = p[NF];        // features 0..15, rows k, k+1
    f1[0] = p[16];  f1[1] = p[NF + 16];   // features 16..31
    a00 = wmma_f32(f0, f0, a00);
    a01 = wmma_f32(f0, f1, a01);
    a10 = wmma_f32(f1, f0, a10);
    a11 = wmma_f32(f1, f1, a11);
  }

  const float inv63 = 1.0f / 63.0f;
  const int row0 = lh * 8;
  float sq = 0.f;
#pragma unroll
  for (int r = 0; r < 8; ++r) {
    float v00 = a00[r] * inv63, v01 = a01[r] * inv63;
    float v10 = a10[r] * inv63, v11 = a11[r] * inv63;
    C[(row0 + r) * ND + l16] = v00;
    C[(row0 + r) * ND + 16 + l16] = v01;
    C[(16 + row0 + r) * ND + l16] = v10;
    C[(16 + row0 + r) * ND + 16 + l16] = v11;
    sq += v00 * v00 + v01 * v01 + v10 * v10 + v11 * v11;
  }
#pragma unroll
  for (int off = 16; off > 0; off >>= 1) sq += __shfl_xor(sq, off, 32);
  if (lane == 0) (tensor ? tt : ss)[t * NN + n] = sq;
}

// ---------------------------------------------------------------------------
// 4) st[t,n,j] = <C_s[t,n], C_t[t,j]> : one wave per pair, lane-strided MACs.
//    grid = 256 blocks, 256 threads (8 waves) -> 2048 pairs.
// ---------------------------------------------------------------------------
__global__ void st_kernel(const float* __restrict__ Cs,
                          const float* __restrict__ Ct,
                          float* __restrict__ st) {
  const int wave = threadIdx.x >> 5, lane = threadIdx.x & 31;
  const int pair = blockIdx.x * 8 + wave;  // t*256 + n*16 + j
  const int j = pair & 15, n = (pair >> 4) & 15, t = pair >> 8;
  const float* A = Cs + (t * NN + n) * (ND * ND);
  const float* Bm = Ct + (t * NN + j) * (ND * ND);
  float s = 0.f;
#pragma unroll 8
  for (int i = lane; i < ND * ND; i += 32) s += A[i] * Bm[i];
#pragma unroll
  for (int off = 16; off > 0; off >>= 1) s += __shfl_xor(s, off, 32);
  if (lane == 0) st[pair] = s;
}

// ---------------------------------------------------------------------------
// 5) L_sca (sum over m,n; mean applied in finalize). grid = 8, 256 threads.
// ---------------------------------------------------------------------------
__global__ void sca_kernel(const float* __restrict__ Es,
                           const float* __restrict__ Et,
                           float* __restrict__ Lsca) {
  const int t = blockIdx.x, mn = threadIdx.x;
  const float* ps = Es + t * 256 + mn;
  const float* pt = Et + t * 256 + mn;
  float sum_s = 0.f, sq_s = 0.f, sum_t = 0.f, sq_t = 0.f;
#pragma unroll 8
  for (int b = 0; b < NB; ++b) {
    float x = ps[b * 2048], y = pt[b * 2048];
    sum_s += x; sq_s += x * x;
    sum_t += y; sq_t += y * y;
  }
  const float var_s = (sq_s - sum_s * sum_s * (1.f / NB)) * (1.f / (NB - 1));
  const float var_t = (sq_t - sum_t * sum_t * (1.f / NB)) * (1.f / (NB - 1));
  float v = var_s - var_t;
  v = v * v * 0.25f;

  __shared__ float red[256];
  red[mn] = v;
  __syncthreads();
  for (int s = 128; s > 0; s >>= 1) {
    if (mn < s) red[mn] += red[mn + s];
    __syncthreads();
  }
  if (mn == 0) Lsca[t] = red[0];
}

// ---------------------------------------------------------------------------
// 6) Gram norms: per-layer ||G||^2 and full-feature ||sum_t G||^2.
//    grid = 3 (type), 256 threads.
// ---------------------------------------------------------------------------
__global__ void norms_kernel(const float* __restrict__ grams,
                             float* __restrict__ nlayer,
                             float* __restrict__ nfull) {
  const int g = blockIdx.x;
  float accL[NL] = {0.f, 0.f, 0.f, 0.f, 0.f, 0.f, 0.f, 0.f};
  float accF = 0.f;
  for (int p = threadIdx.x; p < 64 * 64; p += 256) {
    float sfull = 0.f;
#pragma unroll
    for (int t = 0; t < NL; ++t) {
      float v = grams[(t * 3 + g) * (64 * 64) + p];
      accL[t] += v * v;
      sfull += v;
    }
    accF += sfull * sfull;
  }
  __shared__ float red[256];
#pragma unroll
  for (int t = 0; t < NL; ++t) {
    red[threadIdx.x] = accL[t];
    __syncthreads();
    for (int s = 128; s > 0; s >>= 1) {
      if (threadIdx.x < s) red[threadIdx.x] += red[threadIdx.x + s];
      __syncthreads();
    }
    if (threadIdx.x == 0) nlayer[g * NL + t] = red[0];
    __syncthreads();
  }
  red[threadIdx.x] = accF;
  __syncthreads();
  for (int s = 128; s > 0; s >>= 1) {
    if (threadIdx.x < s) red[threadIdx.x] += red[threadIdx.x + s];
    __syncthreads();
  }
  if (threadIdx.x == 0) nfull[g] = red[0];
}

// ---------------------------------------------------------------------------
// 7) Finalize (scalar): W, L_sfa softmax terms, L_exo, combine. 1 thread.
// ---------------------------------------------------------------------------
__global__ void final_kernel(const float* __restrict__ nlayer,
                             const float* __restrict__ nfull,
                             const float* __restrict__ Lsca,
                             const float* __restrict__ ss,
                             const float* __restrict__ tt,
                             const float* __restrict__ st,
                             float* __restrict__ out) {
  if (threadIdx.x != 0) return;
  const float inv63sq = 1.0f / (63.0f * 63.0f);
  const float invDden = 1.0f / (4.0f * ND * ND);  // 1/4096
  float L_iendo = 0.f;
  for (int t = 0; t < NL; ++t) {
    const float nss = nlayer[0 * NL + t];
    const float ntt = nlayer[1 * NL + t];
    const float nst = nlayer[2 * NL + t];
    const float W =
        (nss - 2.f * nst + ntt) * inv63sq * (1.0f / (4.0f * NF * NF));
    const float lsca = Lsca[t] * (1.0f / (NN * NN));
    float lsfa = 0.f;
    for (int n = 0; n < NN; ++n) {
      const float sn = ss[t * NN + n];
      float pos = 0.f, neg = 0.f;
      for (int j = 0; j < NN; ++j) {
        const float D =
            (sn + tt[t * NN + j] - 2.f * st[(t * NN + n) * NN + j]) * invDden;
        if (j == n) pos = D; else neg += D;
      }
      const float ep = expf(pos);
      lsfa += -logf(ep / (ep + neg + 1e-8f));
    }
    lsfa *= (1.0f / NN);
    L_iendo += W * (0.1f * lsca + 0.1f * lsfa);
  }
  const float fF = (float)(NL * NF);
  const float L_exo =
      (nfull[0] - 2.f * nfull[2] + nfull[1]) * inv63sq * (1.0f / (4.0f * fF * fF));
  out[0] = L_exo + L_iendo * (1.0f / NL);
}

// ---------------------------------------------------------------------------
// Launch. Workspace layout (floats):
//   ZCs 0..262143 | ZCt 262144.. | grams 524288 (24*4096) | Cs 622592 (128k)
//   Ct 753664 | ss 884736 | tt 884864 | st 884992 (2048) | Lsca 887040
//   nlayer 887048 (24) | nfull 887072 (3)   -> ~3.55 MB total
// ---------------------------------------------------------------------------
extern "C" void kernel_launch(void* const* d_in, const int* in_sizes, int n_in,
                              void* d_out, int out_size, void* d_ws,
                              size_t ws_size, hipStream_t stream) {
  const float* Zs = (const float*)d_in[0];
  const float* Es = (const float*)d_in[1];
  const float* Zt = (const float*)d_in[2];
  const float* Et = (const float*)d_in[3];
  float* out = (float*)d_out;
  float* ws = (float*)d_ws;

  float* ZCs = ws + 0;
  float* ZCt = ws + 262144;
  float* GR  = ws + 524288;
  float* Cs  = ws + 622592;
  float* Ct  = ws + 753664;
  float* SS  = ws + 884736;
  float* TT  = ws + 884864;
  float* ST  = ws + 884992;
  float* LS  = ws + 887040;
  float* NLr = ws + 887048;
  float* NFr = ws + 887072;

  const size_t gram_lds = size_t(2) * NB * LROW * sizeof(float);  // 67584 B

  center_kernel<<<16, 256, 0, stream>>>(Zs, Zt, ZCs, ZCt);
  gram_kernel<<<24, 512, gram_lds, stream>>>(ZCs, ZCt, GR);
  cov_kernel<<<256, 32, 0, stream>>>(ZCs, ZCt, Cs, Ct, SS, TT);
  st_kernel<<<256, 256, 0, stream>>>(Cs, Ct, ST);
  sca_kernel<<<8, 256, 0, stream>>>(Es, Et, LS);
  norms_kernel<<<3, 256, 0, stream>>>(GR, NLr, NFr);
  final_kernel<<<1, 32, 0, stream>>>(NLr, NFr, LS, SS, TT, ST, out);
}